// IndexedLinearLayer_1245540516046
// MI455X (gfx1250) — compile-verified
//
#include <hip/hip_runtime.h>
#include <hip/hip_bf16.h>

// Indexed linear layer as a masked dense GEMM over expanded K = NUM_Q * SIZE_IN,
// computed with V_WMMA_F32_16X16X4_F32 (full f32 precision, wave32 WMMA).

#define BATCH    512
#define SIZE_IN  1024
#define SIZE_OUT 256
#define NUM_Q    16

#define KCHUNK   64     // K elements staged per iteration
#define COLS_PB  128    // output columns per block (8 waves * 16)

typedef __attribute__((ext_vector_type(2))) float v2f;
typedef __attribute__((ext_vector_type(8))) float v8f;

// LDS strides padded so float4 stores stay 16B-aligned and B reads are
// conflict-free (lanes 0..15 read 16 consecutive floats of one K-row).
#define A_STRIDE 72     // 64 + pad; 72*4 = 288 bytes (16B multiple)
#define B_STRIDE 136    // 128 + pad; 136*4 = 544 bytes (16B multiple)

__global__ __launch_bounds__(256)
void indexed_linear_wmma_f32(const float* __restrict__ x,
                             const long long* __restrict__ indices,
                             const float* __restrict__ table,
                             const float* __restrict__ bias,
                             float* __restrict__ out) {
    __shared__ float As[16][A_STRIDE];      // masked x tile: [m=batch row][k]
    __shared__ float Bs[KCHUNK][B_STRIDE];  // weight tile:   [k][n]

    const int tid  = threadIdx.x;           // 0..255 (8 wave32s)
    const int lane = tid & 31;
    const int wave = tid >> 5;              // 0..7

    const int bt = blockIdx.x >> 1;         // batch tile (16 rows)
    const int nt = blockIdx.x & 1;          // which 128-col half of SIZE_OUT

    // --- A staging coords: thread loads 4 consecutive features of one row ---
    const int am = tid >> 4;                // 0..15 batch row within tile
    const int ak = (tid & 15) << 2;         // 0,4,...,60 feature offset in chunk
    const int ab = bt * 16 + am;            // global batch row

    // --- WMMA fragment coords (wave32 layouts from CDNA5 ISA) ---
    const int mrow = lane & 15;             // A: M = lane%16 ; B/C: N = lane%16
    const int hi   = lane >> 4;             // 0: lanes 0-15, 1: lanes 16-31
    const int n0   = wave * 16;             // wave's 16-col slice within 128

    v8f acc = {};                           // 16x16 f32 accumulator (8 VGPRs)

    for (int q = 0; q < NUM_Q; ++q) {
        const long long qll = (long long)q;
        for (int kc = 0; kc < SIZE_IN; kc += KCHUNK) {
            // ---- stage masked A: As[m][k] = (idx[b, kc+k]==q) ? x[b, kc+k] : 0
            {
                const int gi = kc + ak;
                const float4 xv = *(const float4*)(x + (size_t)ab * SIZE_IN + gi);
                const long long* ip = indices + (size_t)ab * SIZE_IN + gi;
                float4 av;
                av.x = (ip[0] == qll) ? xv.x : 0.0f;
                av.y = (ip[1] == qll) ? xv.y : 0.0f;
                av.z = (ip[2] == qll) ? xv.z : 0.0f;
                av.w = (ip[3] == qll) ? xv.w : 0.0f;
                *(float4*)&As[am][ak] = av;
            }
            // ---- stage B: Bs[r][c] = table[(q*SIZE_IN + kc + r)*SIZE_OUT + nt*128 + c]
            #pragma unroll
            for (int it = 0; it < 8; ++it) {
                const int f = tid + it * 256;     // float4 id, 0..2047
                const int r = f >> 5;             // K row 0..63
                const int c = (f & 31) << 2;      // col 0..124
                const float4 wv = *(const float4*)(
                    table + (size_t)(q * SIZE_IN + kc + r) * SIZE_OUT + nt * COLS_PB + c);
                *(float4*)&Bs[r][c] = wv;
            }
            // ---- prefetch next chunk's B rows toward L2/WGP$ (global_prefetch_b8)
            if (kc + KCHUNK < SIZE_IN) {
                const int pr = tid >> 2;          // 0..63
                const int pc = (tid & 3) << 5;    // 0,32,64,96
                __builtin_prefetch(
                    table + (size_t)(q * SIZE_IN + kc + KCHUNK + pr) * SIZE_OUT
                          + nt * COLS_PB + pc, 0, 1);
            }
            __syncthreads();

            // ---- 16 x V_WMMA_F32_16X16X4_F32 over the 64-wide K chunk ----
            #pragma unroll
            for (int kb = 0; kb < KCHUNK; kb += 4) {
                v2f a, b;
                // A 16x4 f32 layout: VGPR0 = K{0|2}, VGPR1 = K{1|3} (lane half selects)
                a.x = As[mrow][kb + 2 * hi + 0];
                a.y = As[mrow][kb + 2 * hi + 1];
                // B 4x16 f32 layout: VGPR0 rows K0/K1, VGPR1 rows K2/K3 across lane halves
                b.x = Bs[kb + hi + 0][n0 + mrow];
                b.y = Bs[kb + hi + 2][n0 + mrow];
                acc = __builtin_amdgcn_wmma_f32_16x16x4_f32(
                    /*neg_a=*/false, a, /*neg_b=*/false, b,
                    /*c_mod=*/(short)0, acc, /*reuse_a=*/false, /*reuse_b=*/false);
            }
            __syncthreads();
        }
    }

    // ---- epilogue: C/D layout VGPR r -> M = r + 8*hi, N = lane%16 ----
    const int n = nt * COLS_PB + n0 + mrow;
    const float bv = bias[n];
    #pragma unroll
    for (int r = 0; r < 8; ++r) {
        const int mm = r + 8 * hi;
        out[(size_t)(bt * 16 + mm) * SIZE_OUT + n] = acc[r] + bv;
    }
}

extern "C" void kernel_launch(void* const* d_in, const int* in_sizes, int n_in,
                              void* d_out, int out_size, void* d_ws, size_t ws_size,
                              hipStream_t stream) {
    (void)in_sizes; (void)n_in; (void)out_size; (void)d_ws; (void)ws_size;
    const float*      x     = (const float*)d_in[0];
    const long long*  idx   = (const long long*)d_in[1];   // int64 indices
    const float*      table = (const float*)d_in[2];
    const float*      bias  = (const float*)d_in[3];
    float*            out   = (float*)d_out;

    dim3 grid((BATCH / 16) * (SIZE_OUT / COLS_PB));  // 32 * 2 = 64 blocks
    dim3 block(256);                                  // 8 wave32s
    indexed_linear_wmma_f32<<<grid, block, 0, stream>>>(x, idx, table, bias, out);
}